// EquiformerV3BodyOrderTest_42305427866213
// MI455X (gfx1250) — compile-verified
//
#include <hip/hip_runtime.h>
#include <cmath>

typedef __attribute__((ext_vector_type(2))) float v2f;
typedef __attribute__((ext_vector_type(8))) float v8f;

#define LMAXD 4
#define L2C   25          // (LMAX+1)^2
#define RESB  18
#define RESA  19
#define GRD   342         // RESB*RESA
#define GP    352         // padded to 22*16 grid rows
#define KL    28          // 25 -> padded K for to-grid
#define KG    352         // padded K for from-grid
#define CCH   128
#define NSAMP 6000
#define SB    8           // samples per WG in linear kernels
#define ROWS  240         // 16 * (1+2+3+4+5) padded GEMM rows
#define PI_D  3.14159265358979323846

__device__ __forceinline__ int LDEG(int l) {
  return (l < 1) ? 0 : (l < 4) ? 1 : (l < 9) ? 2 : (l < 16) ? 3 : 4;
}
__device__ __forceinline__ int ZB(int d) { return d * (d + 1) / 2; } // zone base tile

__device__ __forceinline__ v2f ld2(const float* p) {
  v2f r; r.x = p[0]; r.y = p[1]; return r;
}

__device__ __forceinline__ v8f wmma4(v2f a, v2f b, v8f c) {
  // D = A(16x4,f32) * B(4x16,f32) + C(16x16,f32)
  return __builtin_amdgcn_wmma_f32_16x16x4_f32(false, a, false, b, (short)0, c, false, false);
}

// ---------------------------------------------------------------------------
// Kernel A: build SH sampling matrices (Gauss-Legendre x uniform phi) in ws.
// to_grid: [GP][KL] row-major (pads zero);  from_grid: [32][KG] (pads zero).
// ---------------------------------------------------------------------------
__global__ __launch_bounds__(256) void build_sh_kernel(float* __restrict__ to_grid,
                                                       float* __restrict__ from_grid) {
  __shared__ double s_x[RESB], s_w[RESB];
  const int t = threadIdx.x;
  if (t < RESB) {
    const int n = RESB;
    double x = cos(PI_D * (t + 0.75) / (n + 0.5));
    double p0, p1, p2, dp = 1.0;
    for (int it = 0; it < 50; ++it) {
      p0 = 1.0; p1 = x;
      for (int k = 2; k <= n; ++k) { p2 = ((2.0 * k - 1.0) * x * p1 - (k - 1.0) * p0) / k; p0 = p1; p1 = p2; }
      dp = n * (x * p1 - p0) / (x * x - 1.0);
      x -= p1 / dp;
    }
    p0 = 1.0; p1 = x;
    for (int k = 2; k <= n; ++k) { p2 = ((2.0 * k - 1.0) * x * p1 - (k - 1.0) * p0) / k; p0 = p1; p1 = p2; }
    dp = n * (x * p1 - p0) / (x * x - 1.0);
    s_x[t] = x;
    s_w[t] = 2.0 / ((1.0 - x * x) * dp * dp);
  }
  __syncthreads();
  // zero only padding entries (no overlap with real fills below)
  for (int i = t; i < GP * KL; i += 256) {
    int g = i / KL, k = i - g * KL;
    if (g >= GRD || k >= L2C) to_grid[i] = 0.f;
  }
  for (int i = t; i < 32 * KG; i += 256) {
    int l = i / KG, k = i - l * KG;
    if (l >= L2C || k >= GRD) from_grid[i] = 0.f;
  }
  for (int g = t; g < GRD; g += 256) {
    const int b = g / RESA, a = g - b * RESA;
    const double ct = s_x[b];
    const double wq = s_w[b] * (2.0 * PI_D / RESA);
    const double sx = sqrt(fmax(0.0, 1.0 - ct * ct));
    double P[LMAXD + 1][LMAXD + 1];
    P[0][0] = 1.0;
    for (int m = 1; m <= LMAXD; ++m) P[m][m] = -(2.0 * m - 1.0) * sx * P[m - 1][m - 1];
    for (int m = 0; m < LMAXD; ++m) P[m + 1][m] = (2.0 * m + 1.0) * ct * P[m][m];
    for (int m = 0; m <= LMAXD; ++m)
      for (int l = m + 2; l <= LMAXD; ++l)
        P[l][m] = ((2.0 * l - 1.0) * ct * P[l - 1][m] - (l + m - 1.0) * P[l - 2][m]) / (l - m);
    const double phi = a * (2.0 * PI_D / RESA);
    for (int l = 0; l <= LMAXD; ++l) {
      for (int m = 0; m <= l; ++m) {
        double fr = 1.0;
        for (int q = l - m + 1; q <= l + m; ++q) fr *= (double)q;   // (l+m)!/(l-m)!
        const double Nlm = sqrt((2.0 * l + 1.0) / (4.0 * PI_D) / fr);
        if (m == 0) {
          const double y = Nlm * P[l][0];
          to_grid[g * KL + (l * l + l)] = (float)y;
          from_grid[(l * l + l) * KG + g] = (float)(wq * y);
        } else {
          const double base = sqrt(2.0) * Nlm * P[l][m];
          const double yc = base * cos(m * phi);
          const double ys = base * sin(m * phi);
          to_grid[g * KL + (l * l + l + m)] = (float)yc;
          from_grid[(l * l + l + m) * KG + g] = (float)(wq * yc);
          to_grid[g * KL + (l * l + l - m)] = (float)ys;
          from_grid[(l * l + l - m) * KG + g] = (float)(wq * ys);
        }
      }
    }
  }
}

// ---------------------------------------------------------------------------
// Kernel B: merge-layernorm + per-degree linear #1.  8 samples per WG.
// Rows in LDS GEMM matrix are (degree zone, m, sample); zone d starts at row
// 16*ZB(d), holds 8*(2d+1) real rows + 8 zero pad rows.
// ---------------------------------------------------------------------------
__global__ __launch_bounds__(256) void norm_lin1_kernel(
    const float* __restrict__ x, const float* __restrict__ norm_w,
    const float* __restrict__ norm_b, const float* __restrict__ w1,
    const float* __restrict__ b1, float* __restrict__ h1) {
  extern __shared__ float sm[];                       // ROWS * CCH
  const int t = threadIdx.x;
  const int wave = t >> 5, ln = t & 31, hl = ln >> 4, lm = ln & 15;
  const int wg = blockIdx.x;

  for (int i = t; i < ROWS * CCH; i += 256) sm[i] = 0.f;
  __syncthreads();

  { // phase 1: each wave layernorms one sample into the LDS GEMM layout
    const int s = wave;
    const float* xs = x + (size_t)(wg * SB + s) * L2C * CCH;
    float acc = 0.f;
    for (int i = ln; i < CCH; i += 32) acc += xs[i];
    for (int o = 16; o; o >>= 1) acc += __shfl_xor(acc, o, 32);
    const float mean0 = acc * (1.f / CCH);
    float ss = 0.f;
    for (int i = ln; i < L2C * CCH; i += 32) {
      float v = xs[i];
      if (i < CCH) v -= mean0;
      ss += v * v;
    }
    for (int o = 16; o; o >>= 1) ss += __shfl_xor(ss, o, 32);
    const float rs = __frsqrt_rn(ss * (1.f / (L2C * CCH)) + 1e-5f);
    for (int i = ln; i < L2C * CCH; i += 32) {
      const int l = i >> 7, c = i & 127;
      const int d = LDEG(l), m = l - d * d;
      float v = xs[i];
      if (l == 0) v -= mean0;
      v = v * rs * norm_w[d * CCH + c];
      if (l == 0) v += norm_b[c];
      sm[(16 * ZB(d) + m * 8 + s) * CCH + c] = v;
    }
  }
  __syncthreads();

  // phase 2: WMMA GEMM, wave owns output channels [16*wave, 16*wave+16)
  const int co = 16 * wave + lm;
  for (int d = 0; d <= 4; ++d) {
    v2f bf[32];                                        // B-fragment cache
    const float* wd = w1 + ((size_t)d * CCH + co) * CCH;
#pragma unroll
    for (int kk = 0; kk < 32; ++kk) bf[kk] = ld2(wd + 4 * kk + 2 * hl);
    for (int tt = 0; tt <= d; ++tt) {
      const int rbase = 16 * (ZB(d) + tt);
      v8f acc = {};
#pragma unroll
      for (int kk = 0; kk < 32; ++kk) {
        v2f a = ld2(&sm[(rbase + lm) * CCH + 4 * kk + 2 * hl]);
        acc = wmma4(a, bf[kk], acc);
      }
      if (d == 0) {
        const float bb = b1[co];
#pragma unroll
        for (int v = 0; v < 8; ++v) acc[v] += bb;
      }
#pragma unroll
      for (int v = 0; v < 8; ++v) {
        const int rloc = 16 * tt + v + 8 * hl;         // row within zone
        const int m = rloc >> 3, s = rloc & 7;
        if (m < 2 * d + 1) {
          const int l = d * d + m;
          h1[((size_t)(wg * SB + s) * L2C + l) * CCH + co] = acc[v];
        }
      }
    }
  }
}

// ---------------------------------------------------------------------------
// Kernel C: Gaunt self tensor product, fully WMMA, one sample per WG, in place.
// to-grid (K=28) -> square (regs) -> LDS reshape -> from-grid (K streamed).
// ---------------------------------------------------------------------------
__global__ __launch_bounds__(256) void gaunt_kernel(
    const float* __restrict__ to_grid, const float* __restrict__ from_grid,
    const float* __restrict__ w_tp, float* __restrict__ h) {
  extern __shared__ float sm[];                        // hT[CCH][KL] + g2T[8][16][16]
  float* hT = sm;
  float* g2T = sm + CCH * KL;
  const int t = threadIdx.x;
  const int wave = t >> 5, ln = t & 31, hl = ln >> 4, lm = ln & 15;
  float* hn = h + (size_t)blockIdx.x * L2C * CCH;

  for (int i = t; i < CCH * KL; i += 256) hT[i] = 0.f; // zero l-padding
  __syncthreads();
  for (int i = t; i < L2C * CCH; i += 256) {
    const int l = i >> 7, c = i & 127;
    hT[c * KL + l] = hn[i];
  }
  __syncthreads();

  const int cb = 16 * wave;                            // wave's channel base
  float* g2w = g2T + wave * 256;
  v8f acc0 = {}, acc1 = {};
  for (int gt = 0; gt < 22; ++gt) {
    v8f g = {};
#pragma unroll
    for (int kk = 0; kk < 7; ++kk) {                   // K = 28 (l, zero padded)
      v2f a = ld2(&to_grid[(16 * gt + lm) * KL + 4 * kk + 2 * hl]);
      v2f b = ld2(&hT[(cb + lm) * KL + 4 * kk + 2 * hl]);
      g = wmma4(a, b, g);
    }
#pragma unroll
    for (int v = 0; v < 8; ++v) g[v] *= g[v];          // pointwise square on sphere
    // reshape D-frag -> [c][k] in wave-private LDS (contiguous over v)
    float* dst = g2w + lm * 16 + 8 * hl;
#pragma unroll
    for (int v = 0; v < 8; ++v) dst[v] = g[v];
    // from-grid accumulation, K = this 16-row grid chunk
#pragma unroll
    for (int kk = 0; kk < 4; ++kk) {
      v2f b = ld2(&g2w[lm * 16 + 4 * kk + 2 * hl]);
      v2f a0 = ld2(&from_grid[(size_t)lm * KG + 16 * gt + 4 * kk + 2 * hl]);
      v2f a1 = ld2(&from_grid[(size_t)(16 + lm) * KG + 16 * gt + 4 * kk + 2 * hl]);
      acc0 = wmma4(a0, b, acc0);
      acc1 = wmma4(a1, b, acc1);
    }
  }
  // per-degree scale and in-place writeback
#pragma unroll
  for (int v = 0; v < 8; ++v) {
    const int l0 = v + 8 * hl;
    hn[(size_t)l0 * CCH + cb + lm] = acc0[v] * w_tp[LDEG(l0) * CCH + cb + lm];
    const int l1 = 16 + v + 8 * hl;
    if (l1 < L2C)
      hn[(size_t)l1 * CCH + cb + lm] = acc1[v] * w_tp[LDEG(l1) * CCH + cb + lm];
  }
}

// ---------------------------------------------------------------------------
// Kernel D: per-degree linear #2 + bias(l=0) + residual.  Same GEMM as B.
// ---------------------------------------------------------------------------
__global__ __launch_bounds__(256) void lin2_kernel(
    const float* __restrict__ h, const float* __restrict__ x,
    const float* __restrict__ w2, const float* __restrict__ b2,
    float* __restrict__ out) {
  extern __shared__ float sm[];
  const int t = threadIdx.x;
  const int wave = t >> 5, ln = t & 31, hl = ln >> 4, lm = ln & 15;
  const int wg = blockIdx.x;
  for (int i = t; i < ROWS * CCH; i += 256) sm[i] = 0.f;
  __syncthreads();
  for (int i = t; i < SB * L2C * CCH; i += 256) {
    const int c = i & 127, sl = i >> 7;
    const int l = sl % L2C, s = sl / L2C;
    const int d = LDEG(l), m = l - d * d;
    sm[(16 * ZB(d) + m * 8 + s) * CCH + c] =
        h[((size_t)(wg * SB + s) * L2C + l) * CCH + c];
  }
  __syncthreads();

  const int co = 16 * wave + lm;
  for (int d = 0; d <= 4; ++d) {
    v2f bf[32];
    const float* wd = w2 + ((size_t)d * CCH + co) * CCH;
#pragma unroll
    for (int kk = 0; kk < 32; ++kk) bf[kk] = ld2(wd + 4 * kk + 2 * hl);
    for (int tt = 0; tt <= d; ++tt) {
      const int rbase = 16 * (ZB(d) + tt);
      v8f acc = {};
#pragma unroll
      for (int kk = 0; kk < 32; ++kk) {
        v2f a = ld2(&sm[(rbase + lm) * CCH + 4 * kk + 2 * hl]);
        acc = wmma4(a, bf[kk], acc);
      }
      const float bb = (d == 0) ? b2[co] : 0.f;
#pragma unroll
      for (int v = 0; v < 8; ++v) {
        const int rloc = 16 * tt + v + 8 * hl;
        const int m = rloc >> 3, s = rloc & 7;
        if (m < 2 * d + 1) {
          const int l = d * d + m;
          const size_t idx = ((size_t)(wg * SB + s) * L2C + l) * CCH + co;
          out[idx] = acc[v] + bb + x[idx];
        }
      }
    }
  }
}

// ---------------------------------------------------------------------------
extern "C" void kernel_launch(void* const* d_in, const int* in_sizes, int n_in,
                              void* d_out, int out_size, void* d_ws, size_t ws_size,
                              hipStream_t stream) {
  const float* x      = (const float*)d_in[0];
  // d_in[1] = batch (unused by the reference output)
  const float* norm_w = (const float*)d_in[2];
  const float* norm_b = (const float*)d_in[3];
  const float* w1     = (const float*)d_in[4];
  const float* b1     = (const float*)d_in[5];
  const float* w_tp   = (const float*)d_in[6];
  const float* w2     = (const float*)d_in[7];
  const float* b2     = (const float*)d_in[8];

  float* ws        = (float*)d_ws;
  float* to_grid   = ws;                       // GP*KL
  float* from_grid = to_grid + GP * KL;        // 32*KG
  float* hbuf      = from_grid + 32 * KG;      // NSAMP*L2C*CCH (reused in place)

  build_sh_kernel<<<1, 256, 0, stream>>>(to_grid, from_grid);
  norm_lin1_kernel<<<NSAMP / SB, 256, ROWS * CCH * 4, stream>>>(
      x, norm_w, norm_b, w1, b1, hbuf);
  gaunt_kernel<<<NSAMP, 256, (CCH * KL + 8 * 256) * 4, stream>>>(
      to_grid, from_grid, w_tp, hbuf);
  lin2_kernel<<<NSAMP / SB, 256, ROWS * CCH * 4, stream>>>(
      hbuf, x, w2, b2, (float*)d_out);
}